// Encoder_42502996361302
// MI455X (gfx1250) — compile-verified
//
#include <hip/hip_runtime.h>

#define N_NODES 100000
#define N_EDGES 1600000
#define D       128
#define BATCH   50000

typedef float v2f __attribute__((ext_vector_type(2)));
typedef float v8f __attribute__((ext_vector_type(8)));

// ---------------------------------------------------------------------------
// K1: zero the workspace (neighbor-sum accumulator + degree array).
// Must run every call: atomics below accumulate, and ws is only poisoned once.
// ---------------------------------------------------------------------------
__global__ void zero_ws_kernel(float* __restrict__ p, int n) {
    int i = blockIdx.x * blockDim.x + threadIdx.x;
    if (i < n) p[i] = 0.0f;
}

// ---------------------------------------------------------------------------
// K2: edge scatter. One wave32 per edge: each lane loads float4 of the
// neighbor's feature row (L2-resident: features = 51.2MB < 192MB L2) and
// atomically adds into the accumulator row of the edge source. Lane 0 bumps
// the degree counter. Emits global_load_b128 + global_atomic_add_f32.
// ---------------------------------------------------------------------------
__global__ void scatter_edges_kernel(const int* __restrict__ src,
                                     const int* __restrict__ dst,
                                     const float* __restrict__ feat,
                                     float* __restrict__ nsum,
                                     float* __restrict__ deg) {
    int t = blockIdx.x * blockDim.x + threadIdx.x;
    int e = t >> 5;
    int lane = t & 31;
    if (e >= N_EDGES) return;
    int s = src[e];
    int d = dst[e];
    const float4 f = *(const float4*)(feat + (size_t)d * D + lane * 4);
    float* o = nsum + (size_t)s * D + lane * 4;
    atomicAdd(o + 0, f.x);
    atomicAdd(o + 1, f.y);
    atomicAdd(o + 2, f.z);
    atomicAdd(o + 3, f.w);
    if (lane == 0) atomicAdd(deg + s, 1.0f);
}

// ---------------------------------------------------------------------------
// K3: degree-normalize the accumulator in place: mean = sum / max(deg, 1).
// One thread per 4 contiguous floats (same node row).
// ---------------------------------------------------------------------------
__global__ void normalize_kernel(float* __restrict__ nsum,
                                 const float* __restrict__ deg) {
    int t = blockIdx.x * blockDim.x + threadIdx.x;
    int base = t * 4;
    if (base >= N_NODES * D) return;
    int node = base / D;
    float scale = 1.0f / fmaxf(deg[node], 1.0f);
    float4* p = (float4*)(nsum + base);
    float4 v = *p;
    v.x *= scale; v.y *= scale; v.z *= scale; v.w *= scale;
    *p = v;
}

// ---------------------------------------------------------------------------
// K4: fused gather + concat + GEMM + bias + ReLU via V_WMMA_F32_16X16X4_F32.
// One wave32 per 16x16 output tile; 8 waves/block cover all 8 column tiles of
// D=128; grid = BATCH/16 = 3125 row tiles. K=256 accumulated as 64 WMMA steps
// (32 over self features, 32 over neighbor means).
//
// ISA VGPR layouts (CDNA5 ISA, 32-bit 16x4 A / 4x16 B / 16x16 C):
//   lane l: m = l&15, hi = l>>4
//   A vgpr j : K = k0 + 2*hi + j  -> float2 load at &row[k0 + 2*hi]
//   B vgpr j : K = k0 + 2*hi + j, N = l&15 -> float2 at &W[n][k0 + 2*hi]
//   C vgpr v : M = v + 8*hi, N = l&15
// ---------------------------------------------------------------------------
__global__ __launch_bounds__(256) void sage_wmma_gemm_kernel(
    const int*   __restrict__ nodes,
    const float* __restrict__ feat,
    const float* __restrict__ nmean,
    const float* __restrict__ W,      // [D][2*D] row-major
    const float* __restrict__ bias,   // [D]
    float*       __restrict__ out) {  // [BATCH][D]
    const int lane    = threadIdx.x & 31;
    const int wave    = threadIdx.x >> 5;   // column tile 0..7
    const int rowBase = blockIdx.x * 16;
    const int colBase = wave * 16;
    const int nlo     = lane & 15;
    const int hi      = lane >> 4;

    const int node = nodes[rowBase + nlo];
    const float* selfRow  = feat  + (size_t)node * D;
    const float* neighRow = nmean + (size_t)node * D;
    const float* wRow     = W + (size_t)(colBase + nlo) * (2 * D);

    v8f c = {0.f, 0.f, 0.f, 0.f, 0.f, 0.f, 0.f, 0.f};

    // K = 0..127 : self features half of the concat
#pragma unroll
    for (int k0 = 0; k0 < D; k0 += 4) {
        const int kk = k0 + 2 * hi;
        v2f a = *(const v2f*)(selfRow + kk);
        v2f b = *(const v2f*)(wRow + kk);
        c = __builtin_amdgcn_wmma_f32_16x16x4_f32(
            /*neg_a=*/false, a, /*neg_b=*/false, b,
            /*c_mod=*/(short)0, c, /*reuse_a=*/false, /*reuse_b=*/false);
    }
    // K = 128..255 : neighbor-mean half of the concat
#pragma unroll
    for (int k0 = 0; k0 < D; k0 += 4) {
        const int kk = k0 + 2 * hi;
        v2f a = *(const v2f*)(neighRow + kk);
        v2f b = *(const v2f*)(wRow + D + kk);
        c = __builtin_amdgcn_wmma_f32_16x16x4_f32(
            false, a, false, b, (short)0, c, false, false);
    }

    const float bv = bias[colBase + nlo];
#pragma unroll
    for (int v = 0; v < 8; ++v) {
        const int m = rowBase + v + 8 * hi;
        const float r = c[v] + bv;
        out[(size_t)m * D + colBase + nlo] = fmaxf(r, 0.0f);
    }
}

// ---------------------------------------------------------------------------
// Launcher. Inputs (setup_inputs order): nodes[i32], features[f32],
// edge_index[i32 2xE], W[f32 128x256], b[f32 128]. Output: f32 [50000x128].
// Workspace: nsum (N_NODES*D f32) then deg (N_NODES f32), contiguous.
// ---------------------------------------------------------------------------
extern "C" void kernel_launch(void* const* d_in, const int* in_sizes, int n_in,
                              void* d_out, int out_size, void* d_ws, size_t ws_size,
                              hipStream_t stream) {
    const int*   nodes = (const int*)d_in[0];
    const float* feat  = (const float*)d_in[1];
    const int*   edges = (const int*)d_in[2];   // [2][N_EDGES]: row0=src, row1=dst
    const float* W     = (const float*)d_in[3];
    const float* bias  = (const float*)d_in[4];
    float*       out   = (float*)d_out;

    float* nsum = (float*)d_ws;
    float* deg  = nsum + (size_t)N_NODES * D;

    // K1: zero accumulator + degrees (deg is contiguous after nsum)
    const int zn = N_NODES * D + N_NODES;
    zero_ws_kernel<<<(zn + 255) / 256, 256, 0, stream>>>(nsum, zn);

    // K2: scatter-add over edges (one wave32 per edge)
    const int scat_threads = N_EDGES * 32;
    scatter_edges_kernel<<<(scat_threads + 255) / 256, 256, 0, stream>>>(
        edges, edges + N_EDGES, feat, nsum, deg);

    // K3: normalize to mean
    const int norm_threads = (N_NODES * D) / 4;
    normalize_kernel<<<(norm_threads + 255) / 256, 256, 0, stream>>>(nsum, deg);

    // K4: fused gather+concat+GEMM+bias+ReLU via f32 WMMA
    sage_wmma_gemm_kernel<<<BATCH / 16, 256, 0, stream>>>(
        nodes, feat, nsum, W, bias, out);
}